// AttnDecoderRNN_39256001085863
// MI455X (gfx1250) — compile-verified
//
#include <hip/hip_runtime.h>
#include <hip/hip_bf16.h>
#include <math.h>

#define H 256
#define L 512          // MAXLEN
#define V 50257
#define TWOH 512
#define THREEH 768

typedef float v2f __attribute__((ext_vector_type(2)));
typedef float v8f __attribute__((ext_vector_type(8)));

// ---------------- workspace layout (float offsets) ----------------
#define WS_ATTN_IN 0        // 512: [embedded(256), hidden0(256)]
#define WS_LOGITS  512      // 512: attention logits
#define WS_AW      1024     // 512: attention weights
#define WS_X3      1536     // 768: [embedded(256), attn_applied(512)]
#define WS_XCOMB   2304     // 256: relu(comb)
#define WS_GI0     2560     // 1536
#define WS_GH0     4096     // 1536
#define WS_OUT0    5632     // 512: [h00,h01]
#define WS_GI1     6144     // 1536
#define WS_GH1     7680     // 1536
#define WS_OUT1    9216     // 512: [h10,h11]
#define WS_RED     9728     // 2: [max, log(sumexp)]
#define WS_PMAX    9730     // 128 partial maxima
#define WS_PSUM    9858     // 128 partial sums

#define RBLOCKS 128         // blocks for vocab logsumexp stage 1

// ---------------- output layout (floats) ----------------
#define OUT_LOGP 0          // V
#define OUT_HID  V          // 1024
#define OUT_AW   (V + 1024) // 512

__device__ __forceinline__ float sigmoidf_(float x) { return 1.0f / (1.0f + expf(-x)); }

// Embedding lookup + build attn_in = [emb_row, hidden0]; also stash emb row for x3.
__global__ void k_prep(const int* __restrict__ ids, const float* __restrict__ emb,
                       const float* __restrict__ hidden, float* __restrict__ ws) {
    int t = threadIdx.x;                      // 256 threads
    int id = ids[0];
    float e = emb[(size_t)id * H + t];
    ws[WS_ATTN_IN + t]       = e;
    ws[WS_ATTN_IN + H + t]   = hidden[t];     // hidden[0]
    ws[WS_X3 + t]            = e;
}

// Generic wave-per-row matvec: y[r] = act(dot(W[r,:N], x_of(r)) + b[r])
// x_of(r) = x + (r / seg) * xstride   (handles per-direction hidden vectors)
template <int ACT>
__global__ void k_matvec(const float* __restrict__ W, const float* __restrict__ x,
                         int xstride, int seg, const float* __restrict__ b,
                         float* __restrict__ y, int R, int N) {
    int lane = threadIdx.x & 31;
    int wave = threadIdx.x >> 5;
    int row = blockIdx.x * 8 + wave;
    if (row >= R) return;
    const float* w = W + (size_t)row * N;
    const float* xv = x + (size_t)(row / seg) * xstride;
    float acc = 0.0f;
    for (int i = lane; i < N; i += 32) acc = fmaf(w[i], xv[i], acc);
    #pragma unroll
    for (int off = 16; off > 0; off >>= 1) acc += __shfl_down(acc, off);
    if (lane == 0) {
        float v = acc + b[row];
        if (ACT == 1) v = fmaxf(v, 0.0f);
        y[row] = v;
    }
}

// softmax over 512 logits -> ws AW and d_out AW region
__global__ void k_softmax(const float* __restrict__ logits, float* __restrict__ aw_ws,
                          float* __restrict__ aw_out) {
    __shared__ float sm[512];
    int t = threadIdx.x;                      // 512 threads
    float v = logits[t];
    sm[t] = v; __syncthreads();
    for (int s = 256; s > 0; s >>= 1) { if (t < s) sm[t] = fmaxf(sm[t], sm[t + s]); __syncthreads(); }
    float mx = sm[0]; __syncthreads();
    float e = expf(v - mx);
    sm[t] = e; __syncthreads();
    for (int s = 256; s > 0; s >>= 1) { if (t < s) sm[t] += sm[t + s]; __syncthreads(); }
    float r = e / sm[0];
    aw_ws[t] = r;
    aw_out[t] = r;
}

// attn_applied[j] = sum_l aw[l] * ctx[l][j]  (j over 512, coalesced across threads)
__global__ void k_attn_apply(const float* __restrict__ aw, const float* __restrict__ ctx,
                             float* __restrict__ dst) {
    int j = blockIdx.x * blockDim.x + threadIdx.x;   // 0..511
    float acc = 0.0f;
    for (int l = 0; l < L; ++l) acc = fmaf(aw[l], ctx[(size_t)l * TWOH + j], acc);
    dst[j] = acc;
}

// GRU gate combine for 2 directions (blockIdx = dir, 256 threads = H)
__global__ void k_gru_combine(const float* __restrict__ gi, const float* __restrict__ gh,
                              const float* __restrict__ h, float* __restrict__ out_ws,
                              float* __restrict__ out_hid) {
    int d = blockIdx.x, i = threadIdx.x;
    const float* gid = gi + d * THREEH;
    const float* ghd = gh + d * THREEH;
    float r = sigmoidf_(gid[i]         + ghd[i]);
    float z = sigmoidf_(gid[H + i]     + ghd[H + i]);
    float n = tanhf   (gid[2 * H + i]  + r * ghd[2 * H + i]);
    float hn = (1.0f - z) * n + z * h[d * H + i];
    out_ws[d * H + i]  = hn;
    out_hid[d * H + i] = hn;
}

// Vocab projection via v_wmma_f32_16x16x4_f32.
// A = 16 rows of out_W (per-wave tile), B = x chunk replicated across all 16 columns,
// so every column of D holds y[base..base+15]. 128 chained WMMAs sweep K=512.
#define NGROUPS ((V + 15) / 16)   // 3142
__global__ void k_out_wmma(const float* __restrict__ Wv, const float* __restrict__ x,
                           const float* __restrict__ bias, float* __restrict__ logits) {
    __shared__ float xs[TWOH];
    for (int i = threadIdx.x; i < TWOH; i += blockDim.x) xs[i] = x[i];
    __syncthreads();

    int lane = threadIdx.x & 31;
    int wave = threadIdx.x >> 5;
    int g = blockIdx.x * 8 + wave;
    if (g >= NGROUPS) return;                 // wave-uniform: EXEC all-1s inside

    int base = g * 16;
    int m = lane & 15;
    int row = base + m; if (row > V - 1) row = V - 1;   // clamp (no divergence)
    int koff = (lane >> 4) * 2;               // half-wave K offset (ISA A layout)
    const float* wrow = Wv + (size_t)row * TWOH + koff;

    v8f c = {};
    #pragma unroll 4
    for (int k = 0; k < TWOH; k += 4) {
        v2f a, b;
        a.x = wrow[k];
        a.y = wrow[k + 1];
        b.x = xs[k + koff];
        b.y = xs[k + koff + 1];
        c = __builtin_amdgcn_wmma_f32_16x16x4_f32(
                /*neg_a=*/false, a, /*neg_b=*/false, b,
                /*c_mod=*/(short)0, c, /*reuse_a=*/false, /*reuse_b=*/false);
    }

    // D layout: VGPR j -> row M=j (lanes 0-15) / M=8+j (lanes 16-31); column n=0|16.
    if (lane == 0 || lane == 16) {
        int rb = base + (lane >> 4) * 8;
        if (base + 15 < V) {
            // full group (3141 of 3142 waves): straight unguarded stores
            #pragma unroll
            for (int j = 0; j < 8; ++j) logits[rb + j] = c[j] + bias[rb + j];
        } else {
            #pragma unroll
            for (int j = 0; j < 8; ++j) {
                int r = rb + j;
                if (r < V) logits[r] = c[j] + bias[r];
            }
        }
    }
}

// ---- parallel log-sum-exp over V logits (stage 1: per-block partials) ----
__global__ void k_vred1(const float* __restrict__ logits, float* __restrict__ pm,
                        float* __restrict__ ps) {
    __shared__ float sm[256];
    int t = threadIdx.x;
    int stride = RBLOCKS * 256;
    int i0 = blockIdx.x * 256 + t;
    float m = -INFINITY;
    for (int i = i0; i < V; i += stride) m = fmaxf(m, logits[i]);
    sm[t] = m; __syncthreads();
    for (int s = 128; s > 0; s >>= 1) { if (t < s) sm[t] = fmaxf(sm[t], sm[t + s]); __syncthreads(); }
    float mx = sm[0]; __syncthreads();
    float sum = 0.0f;
    for (int i = i0; i < V; i += stride) sum += expf(logits[i] - mx);
    sm[t] = sum; __syncthreads();
    for (int s = 128; s > 0; s >>= 1) { if (t < s) sm[t] += sm[t + s]; __syncthreads(); }
    if (t == 0) { pm[blockIdx.x] = mx; ps[blockIdx.x] = sm[0]; }
}

// ---- stage 2: merge 128 partials -> red[0]=max, red[1]=log(sumexp) ----
__global__ void k_vred2(const float* __restrict__ pm, const float* __restrict__ ps,
                        float* __restrict__ red) {
    __shared__ float sm[RBLOCKS];
    int t = threadIdx.x;                      // 128 threads
    float m = pm[t];
    sm[t] = m; __syncthreads();
    for (int s = RBLOCKS / 2; s > 0; s >>= 1) { if (t < s) sm[t] = fmaxf(sm[t], sm[t + s]); __syncthreads(); }
    float mx = sm[0]; __syncthreads();
    float su = ps[t] * expf(m - mx);          // pm=-inf => ps=0 => contributes 0
    sm[t] = su; __syncthreads();
    for (int s = RBLOCKS / 2; s > 0; s >>= 1) { if (t < s) sm[t] += sm[t + s]; __syncthreads(); }
    if (t == 0) { red[0] = mx; red[1] = logf(sm[0]); }
}

__global__ void k_logp(float* __restrict__ logits, const float* __restrict__ red) {
    int i = blockIdx.x * blockDim.x + threadIdx.x;
    if (i < V) logits[i] = logits[i] - red[0] - red[1];
}

extern "C" void kernel_launch(void* const* d_in, const int* in_sizes, int n_in,
                              void* d_out, int out_size, void* d_ws, size_t ws_size,
                              hipStream_t stream) {
    const int*   ids    = (const int*)  d_in[0];
    const float* hidden = (const float*)d_in[1];   // (4,1,256)
    const float* ctx    = (const float*)d_in[2];   // (512,512)
    const float* emb    = (const float*)d_in[3];   // (V,256)
    const float* attn_W = (const float*)d_in[4];   // (512,512)
    const float* attn_b = (const float*)d_in[5];
    const float* comb_W = (const float*)d_in[6];   // (256,768)
    const float* comb_b = (const float*)d_in[7];
    const float* w_ih0  = (const float*)d_in[8];   // (2,768,256)
    const float* w_hh0  = (const float*)d_in[9];   // (2,768,256)
    const float* b_ih0  = (const float*)d_in[10];  // (2,768)
    const float* b_hh0  = (const float*)d_in[11];
    const float* w_ih1  = (const float*)d_in[12];  // (2,768,512)
    const float* w_hh1  = (const float*)d_in[13];  // (2,768,256)
    const float* b_ih1  = (const float*)d_in[14];
    const float* b_hh1  = (const float*)d_in[15];
    const float* out_W  = (const float*)d_in[16];  // (V,512)
    const float* out_b  = (const float*)d_in[17];

    float* ws  = (float*)d_ws;
    float* out = (float*)d_out;

    // 1) embed + attn_in
    k_prep<<<1, 256, 0, stream>>>(ids, emb, hidden, ws);
    // 2) attention logits (512 rows x 512)
    k_matvec<0><<<64, 256, 0, stream>>>(attn_W, ws + WS_ATTN_IN, 0, THREEH, attn_b,
                                        ws + WS_LOGITS, L, TWOH);
    // 3) softmax -> attn_weights (ws + output)
    k_softmax<<<1, 512, 0, stream>>>(ws + WS_LOGITS, ws + WS_AW, out + OUT_AW);
    // 4) attn_applied into x3[256..767]
    k_attn_apply<<<2, 256, 0, stream>>>(ws + WS_AW, ctx, ws + WS_X3 + H);
    // 5) x = relu(comb_W @ x3 + comb_b)  (256 rows x 768)
    k_matvec<1><<<32, 256, 0, stream>>>(comb_W, ws + WS_X3, 0, THREEH, comb_b,
                                        ws + WS_XCOMB, H, THREEH);
    // 6) GRU layer 0 gates: gi (x same both dirs), gh (hidden[0|1])
    k_matvec<0><<<192, 256, 0, stream>>>(w_ih0, ws + WS_XCOMB, 0, THREEH, b_ih0,
                                         ws + WS_GI0, 2 * THREEH, H);
    k_matvec<0><<<192, 256, 0, stream>>>(w_hh0, hidden, H, THREEH, b_hh0,
                                         ws + WS_GH0, 2 * THREEH, H);
    k_gru_combine<<<2, 256, 0, stream>>>(ws + WS_GI0, ws + WS_GH0, hidden,
                                         ws + WS_OUT0, out + OUT_HID);
    // 7) GRU layer 1 gates: input = out0 (512), gh on hidden[2|3]
    k_matvec<0><<<192, 256, 0, stream>>>(w_ih1, ws + WS_OUT0, 0, THREEH, b_ih1,
                                         ws + WS_GI1, 2 * THREEH, TWOH);
    k_matvec<0><<<192, 256, 0, stream>>>(w_hh1, hidden + 2 * H, H, THREEH, b_hh1,
                                         ws + WS_GH1, 2 * THREEH, H);
    k_gru_combine<<<2, 256, 0, stream>>>(ws + WS_GI1, ws + WS_GH1, hidden + 2 * H,
                                         ws + WS_OUT1, out + OUT_HID + 2 * H);
    // 8) vocab projection via WMMA (103 MB stream -> bandwidth bound)
    k_out_wmma<<<(NGROUPS + 7) / 8, 256, 0, stream>>>(out_W, ws + WS_OUT1, out_b,
                                                      out + OUT_LOGP);
    // 9) log-softmax in place (parallel two-stage logsumexp)
    k_vred1<<<RBLOCKS, 256, 0, stream>>>(out + OUT_LOGP, ws + WS_PMAX, ws + WS_PSUM);
    k_vred2<<<1, RBLOCKS, 0, stream>>>(ws + WS_PMAX, ws + WS_PSUM, ws + WS_RED);
    k_logp<<<(V + 255) / 256, 256, 0, stream>>>(out + OUT_LOGP, ws + WS_RED);
}